// HMM_43439299232436
// MI455X (gfx1250) — compile-verified
//
#include <hip/hip_runtime.h>
#include <math.h>

// Problem constants (match reference: B=128, S=8192, K=16, leak=0.1)
#define KS 16
#define BB 128
#define SS 8192
#define NC 32            // chunks per sequence
#define CL (SS / NC)     // 256 operators per chunk

#define LEAKF 0.1f
#define LOG2E_F 1.4426950408889634f
#define LN2_F 0.6931471805599453f
#define HALF_LOG_2PI_F 0.9189385332046727f

typedef float v2f __attribute__((ext_vector_type(2)));
typedef float v8f __attribute__((ext_vector_type(8)));

// wave32 lane XOR exchange via ds_swizzle (group-of-32 mode, and_mask=0x1f)
template <int MASK>
__device__ __forceinline__ float swz_xor(float v) {
  return __int_as_float(
      __builtin_amdgcn_ds_swizzle(__float_as_int(v), (MASK << 10) | 0x1f));
}

// ---------------------------------------------------------------------------
// Phase A: one wave per (batch, chunk). Build the 16x16 exp-domain chunk
// propagation matrix with a scalar log2 shift, exploiting T = a*I + b*11^T.
// Lane l: column j = l&15, rows (l>>4)*8 .. +7 in g[0..7].
// ---------------------------------------------------------------------------
__global__ __launch_bounds__(128) void hmm_chunk_ops(
    const float* __restrict__ obvs, const float* __restrict__ mu,
    const float* __restrict__ log_sigma, const float* __restrict__ prior_logits,
    float* __restrict__ wsM, float* __restrict__ wsS) {
  __shared__ float sobs[4 * CL];
  const int lane = threadIdx.x & 31;
  const int wid = threadIdx.x >> 5;              // 0..3
  const int task = blockIdx.x * 4 + wid;         // 0..4095
  const int b = task / NC;
  const int c = task % NC;
  const int j = lane & 15;
  const int hi = lane >> 4;

  // Stage this wave's 256 observations into LDS (coalesced).
  const float* src = obvs + (size_t)b * SS + (size_t)c * CL;
  float* sob = sobs + wid * CL;
  for (int t = lane; t < CL; t += 32) sob[t] = src[t];
  __syncthreads();

  // Per-state emission constants (base-2 domain).
  const float ls = log_sigma[j];
  const float muj = mu[j];
  const float inv_sig = __builtin_amdgcn_exp2f(-ls * LOG2E_F);
  const float n0 = -(ls + HALF_LOG_2PI_F) * LOG2E_F;
  const float n1 = -0.5f * LOG2E_F;
  const float bcoef = LEAKF / (float)(KS - 1);
  const float acoef = (1.0f - LEAKF) - bcoef;

  float g[8];
  float shift2 = 0.0f;
  int tstart;
  if (c == 0) {
    // Fold carry0 = softmax(prior) * em_0 into a diagonal initial matrix.
    const float x0 = sob[0];
    const float z0 = (x0 - muj) * inv_sig;
    const float em0 = __builtin_amdgcn_exp2f(fmaf(z0 * z0, n1, n0));
    const float q = __builtin_amdgcn_exp2f(prior_logits[j] * LOG2E_F);
    float Q = q;
    Q += swz_xor<1>(Q); Q += swz_xor<2>(Q); Q += swz_xor<4>(Q); Q += swz_xor<8>(Q);
    const float v0 = q * em0 * __builtin_amdgcn_rcpf(Q);
#pragma unroll
    for (int k = 0; k < 8; ++k) g[k] = ((hi * 8 + k) == j) ? v0 : 0.0f;
    tstart = 1;
  } else {
#pragma unroll
    for (int k = 0; k < 8; ++k) g[k] = ((hi * 8 + k) == j) ? 1.0f : 0.0f;
    tstart = 0;
  }

#pragma unroll 4
  for (int t = tstart; t < CL; ++t) {
    const float x = sob[t];
    const float z = (x - muj) * inv_sig;
    const float em = __builtin_amdgcn_exp2f(fmaf(z * z, n1, n0));
    // Row sums over the 16 columns (butterfly within each 16-lane group).
    float r[8];
#pragma unroll
    for (int k = 0; k < 8; ++k) r[k] = g[k];
#pragma unroll
    for (int k = 0; k < 8; ++k) r[k] += swz_xor<1>(r[k]);
#pragma unroll
    for (int k = 0; k < 8; ++k) r[k] += swz_xor<2>(r[k]);
#pragma unroll
    for (int k = 0; k < 8; ++k) r[k] += swz_xor<4>(r[k]);
#pragma unroll
    for (int k = 0; k < 8; ++k) r[k] += swz_xor<8>(r[k]);
    // G' = em_j * (a*G + b*rowsum)
#pragma unroll
    for (int k = 0; k < 8; ++k) g[k] = em * fmaf(bcoef, r[k], acoef * g[k]);

    if ((t & 15) == 15) {  // periodic exp-domain renormalization
      float m = g[0];
#pragma unroll
      for (int k = 1; k < 8; ++k) m = fmaxf(m, g[k]);
      m = fmaxf(m, swz_xor<1>(m));
      m = fmaxf(m, swz_xor<2>(m));
      m = fmaxf(m, swz_xor<4>(m));
      m = fmaxf(m, swz_xor<8>(m));
      m = fmaxf(m, swz_xor<16>(m));
      m = fmaxf(m, 1e-30f);
      const float rm = __builtin_amdgcn_rcpf(m);
#pragma unroll
      for (int k = 0; k < 8; ++k) g[k] *= rm;
      shift2 += __builtin_amdgcn_logf(m);  // v_log_f32 is log2
    }
  }

  float* out = wsM + (size_t)task * 256;  // row-major 16x16
#pragma unroll
  for (int k = 0; k < 8; ++k) out[(hi * 8 + k) * 16 + j] = g[k];
  if (lane == 0) wsS[task] = shift2;
}

// ---------------------------------------------------------------------------
// Phase B helper: one wave computes D = L x R (16x16x16) with 4 chained
// V_WMMA_F32_16X16X4_F32, renormalizes by the max, accumulates log2 shift.
// ---------------------------------------------------------------------------
__device__ __forceinline__ void mat_product_wave(const float* __restrict__ Lm,
                                                 const float* __restrict__ Rm,
                                                 float* __restrict__ Dm,
                                                 float shL, float shR,
                                                 float* __restrict__ shOut,
                                                 int lane) {
  const int lo = lane & 15;
  const int hi = lane >> 4;
  v8f acc = {};
#pragma unroll
  for (int kc = 0; kc < 4; ++kc) {
    const int k0 = 4 * kc + (hi ? 2 : 0);
    const int k1 = 4 * kc + (hi ? 3 : 1);
    v2f a, bb;
    a.x = Lm[lo * 16 + k0];   // A-matrix 16x4 layout: M=lane%16, K split per half
    a.y = Lm[lo * 16 + k1];
    bb.x = Rm[k0 * 16 + lo];  // B-matrix 4x16 layout: rows striped across halves
    bb.y = Rm[k1 * 16 + lo];
    acc = __builtin_amdgcn_wmma_f32_16x16x4_f32(false, a, false, bb, (short)0,
                                                acc, false, false);
  }
  float m = acc[0];
#pragma unroll
  for (int k = 1; k < 8; ++k) m = fmaxf(m, acc[k]);
  m = fmaxf(m, swz_xor<1>(m));
  m = fmaxf(m, swz_xor<2>(m));
  m = fmaxf(m, swz_xor<4>(m));
  m = fmaxf(m, swz_xor<8>(m));
  m = fmaxf(m, swz_xor<16>(m));
  m = fmaxf(m, 1e-30f);
  const float rm = __builtin_amdgcn_rcpf(m);
#pragma unroll
  for (int k = 0; k < 8; ++k) Dm[(hi * 8 + k) * 16 + lo] = acc[k] * rm;
  if (lane == 0) *shOut = shL + shR + __builtin_amdgcn_logf(m);
}

// ---------------------------------------------------------------------------
// Phase B: one block per batch; 16 waves reduce 32 chunk matrices by a
// binary tree of WMMA matmuls in LDS, then emit per-batch log-likelihood.
// ---------------------------------------------------------------------------
__global__ __launch_bounds__(512) void hmm_tree_combine(
    const float* __restrict__ wsM, const float* __restrict__ wsS,
    float* __restrict__ wsR) {
  __shared__ float buf0[32 * 256];
  __shared__ float buf1[16 * 256];
  __shared__ float shA[32];
  __shared__ float shB[16];
  const int b = blockIdx.x;
  const int tid = threadIdx.x;
  const int lane = tid & 31;
  const int w = tid >> 5;  // 0..15

  const float* gsrc = wsM + (size_t)b * NC * 256;
  for (int i = tid; i < 32 * 256; i += 512) buf0[i] = gsrc[i];
  if (tid < 32) shA[tid] = wsS[b * NC + tid];
  __syncthreads();

  if (w < 16)
    mat_product_wave(buf0 + (2 * w) * 256, buf0 + (2 * w + 1) * 256,
                     buf1 + w * 256, shA[2 * w], shA[2 * w + 1], &shB[w], lane);
  __syncthreads();
  if (w < 8)
    mat_product_wave(buf1 + (2 * w) * 256, buf1 + (2 * w + 1) * 256,
                     buf0 + w * 256, shB[2 * w], shB[2 * w + 1], &shA[w], lane);
  __syncthreads();
  if (w < 4)
    mat_product_wave(buf0 + (2 * w) * 256, buf0 + (2 * w + 1) * 256,
                     buf1 + w * 256, shA[2 * w], shA[2 * w + 1], &shB[w], lane);
  __syncthreads();
  if (w < 2)
    mat_product_wave(buf1 + (2 * w) * 256, buf1 + (2 * w + 1) * 256,
                     buf0 + w * 256, shB[2 * w], shB[2 * w + 1], &shA[w], lane);
  __syncthreads();
  if (w < 1)
    mat_product_wave(buf0, buf0 + 256, buf1, shA[0], shA[1], &shB[0], lane);
  __syncthreads();

  // Final product P includes diag(carry0): answer_b = ln2*(shift + log2 sum P).
  if (tid < 32) {
    float s = 0.0f;
#pragma unroll
    for (int k = 0; k < 8; ++k) s += buf1[tid * 8 + k];
    s += swz_xor<1>(s); s += swz_xor<2>(s); s += swz_xor<4>(s);
    s += swz_xor<8>(s); s += swz_xor<16>(s);
    if (tid == 0) wsR[b] = (shB[0] + __builtin_amdgcn_logf(s)) * LN2_F;
  }
}

// ---------------------------------------------------------------------------
// Phase C: deterministic fixed-order reduction of the 128 per-batch results.
// ---------------------------------------------------------------------------
__global__ __launch_bounds__(128) void hmm_final_sum(
    const float* __restrict__ wsR, float* __restrict__ out) {
  __shared__ float part[4];
  const int tid = threadIdx.x;
  float s = wsR[tid];
  s += swz_xor<1>(s); s += swz_xor<2>(s); s += swz_xor<4>(s);
  s += swz_xor<8>(s); s += swz_xor<16>(s);
  if ((tid & 31) == 0) part[tid >> 5] = s;
  __syncthreads();
  if (tid == 0) out[0] = (part[0] + part[1]) + (part[2] + part[3]);
}

extern "C" void kernel_launch(void* const* d_in, const int* in_sizes, int n_in,
                              void* d_out, int out_size, void* d_ws,
                              size_t ws_size, hipStream_t stream) {
  (void)in_sizes; (void)n_in; (void)out_size; (void)ws_size;
  const float* obvs = (const float*)d_in[0];
  const float* mu = (const float*)d_in[1];
  const float* log_sigma = (const float*)d_in[2];
  const float* prior_logits = (const float*)d_in[3];

  // Workspace layout: [B*NC 16x16 matrices][B*NC shifts][B per-batch results]
  float* wsM = (float*)d_ws;                       // 128*32*256 floats
  float* wsS = wsM + (size_t)BB * NC * 256;        // 4096 floats
  float* wsR = wsS + (size_t)BB * NC;              // 128 floats (~4.2 MB total)

  hmm_chunk_ops<<<(BB * NC) / 4, 128, 0, stream>>>(obvs, mu, log_sigma,
                                                   prior_logits, wsM, wsS);
  hmm_tree_combine<<<BB, 512, 0, stream>>>(wsM, wsS, wsR);
  hmm_final_sum<<<1, 128, 0, stream>>>(wsR, (float*)d_out);
}